// GumbelMemoryModel_35270271435222
// MI455X (gfx1250) — compile-verified
//
#include <hip/hip_runtime.h>
#include <hip/hip_bf16.h>

// ---------------------------------------------------------------------------
// GumbelMemoryModel on MI455X (gfx1250), fp32 throughout using
// V_WMMA_F32_16X16X4_F32 for all GEMMs (exact fp32 => top-k ranking safe).
// Round 2: k-outer / multi-accumulator GEMM loops so one A fragment feeds
// 8 (GEMM1) / 4 (GEMM2) / 2 (out GEMM) independent WMMAs per k-step.
// ---------------------------------------------------------------------------

constexpr int kD     = 512;
constexpr int kD2    = 1024;
constexpr int kTF    = 2049;   // T_FULL
constexpr int kTC    = 2048;   // context length
constexpr int kB     = 64;
constexpr int kV     = 50257;
constexpr int kK     = 3;
constexpr int kSlots = 16;
constexpr int HS     = 520;    // LDS row stride for h / x tile (512 + pad)
constexpr int AS     = 1032;   // LDS row stride for activation tile (1024 + pad)

typedef float v2f __attribute__((ext_vector_type(2)));
typedef float v8f __attribute__((ext_vector_type(8)));

static __device__ inline v8f wmma4(v2f a, v2f b, v8f c) {
  // D(16x16,f32) = A(16x4,f32) * B(4x16,f32) + C
  return __builtin_amdgcn_wmma_f32_16x16x4_f32(
      /*neg_a=*/false, a, /*neg_b=*/false, b,
      /*c_mod=*/(short)0, c, /*reuse_a=*/false, /*reuse_b=*/false);
}

// ---------------------------------------------------------------------------
// Kernel 1: per 16-token tile: h = embed[seq]; act = relu(h@W1+b1);
// x = act@W2+b2+h (in-place into h tile); LayerNorm; gate = hid.Wg + bg.
// Stores gate for t<2048 and the full hidden row for t==2048 (query token).
// ---------------------------------------------------------------------------
__global__ __launch_bounds__(256) void ffn_ln_gate(
    const int* __restrict__ seq, const float* __restrict__ embed,
    const float* __restrict__ W1, const float* __restrict__ b1,
    const float* __restrict__ W2, const float* __restrict__ b2,
    const float* __restrict__ gamma, const float* __restrict__ beta,
    const float* __restrict__ Wg, const float* __restrict__ bg,
    float* __restrict__ gate, float* __restrict__ hidden_last)
{
  extern __shared__ float smem[];
  float* hbuf = smem;            // 16 * HS floats (h, later x)
  float* act  = smem + 16 * HS;  // 16 * AS floats

  const int tid   = threadIdx.x;
  const int lane  = tid & 31;
  const int wave  = tid >> 5;
  const int row16 = lane & 15;
  const int koff  = (lane >> 4) << 1;   // 0 or 2: K sub-offset of A/B frags
  const int moff  = (lane >> 4) << 3;   // C layout: M = r + 8*(lane>=16)
  const int tokBase = blockIdx.x * 16;

  // Phase 0: gather 16 embedding rows into LDS
  for (int i = tid; i < 16 * kD; i += 256) {
    int r = i / kD, c = i % kD;
    int v = seq[tokBase + r];
    hbuf[r * HS + c] = embed[(long)v * kD + c];
  }
  __syncthreads();

  // Phase 1: act = relu(h @ W1 + b1); wave owns cols [wave*128, wave*128+128)
  {
    const int colBase = wave * 128 + row16;
    v8f acc[8];
#pragma unroll
    for (int t = 0; t < 8; ++t)
      acc[t] = (v8f){0.f, 0.f, 0.f, 0.f, 0.f, 0.f, 0.f, 0.f};
    for (int k = 0; k < kD; k += 4) {
      v2f a;                                        // one A frag per k-step
      a.x = hbuf[row16 * HS + k + koff];
      a.y = hbuf[row16 * HS + k + koff + 1];
      const float* w1r0 = &W1[(k + koff) * kD2 + colBase];
      const float* w1r1 = &W1[(k + koff + 1) * kD2 + colBase];
#pragma unroll
      for (int t = 0; t < 8; ++t) {                 // 8 independent WMMAs
        v2f bb;
        bb.x = w1r0[t * 16];
        bb.y = w1r1[t * 16];
        acc[t] = wmma4(a, bb, acc[t]);
      }
    }
#pragma unroll
    for (int t = 0; t < 8; ++t) {
      const int col = colBase + t * 16;
      const float bias = b1[col];
#pragma unroll
      for (int r = 0; r < 8; ++r) {
        float v = acc[t][r] + bias;
        act[(r + moff) * AS + col] = v > 0.f ? v : 0.f;
      }
    }
  }
  __syncthreads();

  // Phase 2: x = act @ W2 + b2 + h, written in place into hbuf
  {
    const int colBase = wave * 64 + row16;
    v8f acc[4];
#pragma unroll
    for (int t = 0; t < 4; ++t)
      acc[t] = (v8f){0.f, 0.f, 0.f, 0.f, 0.f, 0.f, 0.f, 0.f};
    for (int k = 0; k < kD2; k += 4) {
      v2f a;
      a.x = act[row16 * AS + k + koff];
      a.y = act[row16 * AS + k + koff + 1];
      const float* w2r0 = &W2[(k + koff) * kD + colBase];
      const float* w2r1 = &W2[(k + koff + 1) * kD + colBase];
#pragma unroll
      for (int t = 0; t < 4; ++t) {
        v2f bb;
        bb.x = w2r0[t * 16];
        bb.y = w2r1[t * 16];
        acc[t] = wmma4(a, bb, acc[t]);
      }
    }
#pragma unroll
    for (int t = 0; t < 4; ++t) {
      const int col = colBase + t * 16;
      const float bias = b2[col];
#pragma unroll
      for (int r = 0; r < 8; ++r)
        hbuf[(r + moff) * HS + col] += acc[t][r] + bias;  // x = h + ff
    }
  }
  __syncthreads();

  // Phase 3: LayerNorm + gate dot; each wave handles 2 token rows
  for (int rr = wave * 2; rr < wave * 2 + 2; ++rr) {
    const int tok = tokBase + rr;
    const int b = tok / kTF;
    const int t = tok - b * kTF;
    float s = 0.f, sq = 0.f;
    for (int c = lane; c < kD; c += 32) {
      float x = hbuf[rr * HS + c];
      s += x; sq += x * x;
    }
#pragma unroll
    for (int m = 16; m > 0; m >>= 1) {
      s  += __shfl_xor(s,  m, 32);
      sq += __shfl_xor(sq, m, 32);
    }
    const float mu  = s  * (1.f / (float)kD);
    const float var = sq * (1.f / (float)kD) - mu * mu;
    const float rs  = rsqrtf(var + 1e-5f);
    float gd = 0.f;
    for (int c = lane; c < kD; c += 32) {
      float v = (hbuf[rr * HS + c] - mu) * rs * gamma[c] + beta[c];
      gd += v * Wg[c];
      if (t == kTC) hidden_last[b * kD + c] = v;   // query-token hidden
    }
#pragma unroll
    for (int m = 16; m > 0; m >>= 1) gd += __shfl_xor(gd, m, 32);
    if (lane == 0 && t < kTC) gate[b * kTC + t] = gd + bg[0];
  }
}

// ---------------------------------------------------------------------------
// Kernel 2: per-batch top-3 of (gate + gumbel) / temperature over T=2048
// ---------------------------------------------------------------------------
__global__ __launch_bounds__(256) void topk_kernel(
    const float* __restrict__ gate, const float* __restrict__ gumbel,
    const float* __restrict__ temperature, int* __restrict__ topidx)
{
  __shared__ float lv[256 * 3];
  __shared__ int   li[256 * 3];
  const int b = blockIdx.x, tid = threadIdx.x;
  const float invT = 1.f / temperature[0];
  float v0 = -__builtin_inff(), v1 = v0, v2 = v0;
  int   i0 = -1, i1 = -1, i2 = -1;
  for (int t = tid; t < kTC; t += 256) {
    float p = (gate[b * kTC + t] + gumbel[b * kTC + t]) * invT;
    if (p > v0)      { v2 = v1; i2 = i1; v1 = v0; i1 = i0; v0 = p; i0 = t; }
    else if (p > v1) { v2 = v1; i2 = i1; v1 = p;  i1 = t; }
    else if (p > v2) { v2 = p;  i2 = t; }
  }
  lv[tid * 3 + 0] = v0; li[tid * 3 + 0] = i0;
  lv[tid * 3 + 1] = v1; li[tid * 3 + 1] = i1;
  lv[tid * 3 + 2] = v2; li[tid * 3 + 2] = i2;
  __syncthreads();
  if (tid == 0) {
    for (int j = 0; j < kK; ++j) {
      float best = -__builtin_inff(); int bi = kTC, slot = -1;
      for (int e = 0; e < 256 * 3; ++e) {
        int ti = li[e];
        if (ti < 0) continue;
        float v = lv[e];
        if (v > best || (v == best && ti < bi)) { best = v; bi = ti; slot = e; }
      }
      topidx[b * kK + j] = bi;
      li[slot] = -1;
    }
  }
}

// ---------------------------------------------------------------------------
// Kernel 3: recompute hidden rows for the 3 selected tokens per batch
// (192 tokens total — negligible FLOPs; simple block GEMV + block LayerNorm)
// ---------------------------------------------------------------------------
__global__ __launch_bounds__(256) void recompute_hidden(
    const int* __restrict__ seq, const float* __restrict__ embed,
    const float* __restrict__ W1, const float* __restrict__ b1,
    const float* __restrict__ W2, const float* __restrict__ b2,
    const float* __restrict__ gamma, const float* __restrict__ beta,
    const int* __restrict__ topidx, float* __restrict__ hid_top)
{
  __shared__ float hrow[kD];
  __shared__ float mid[kD2];
  __shared__ float xr[kD];
  __shared__ float red[256];
  const int b = blockIdx.x / kK, j = blockIdx.x % kK, tid = threadIdx.x;
  const int t = topidx[b * kK + j];
  const int v = seq[b * kTF + t];
  for (int c = tid; c < kD; c += 256) hrow[c] = embed[(long)v * kD + c];
  __syncthreads();
  for (int m = tid; m < kD2; m += 256) {
    float s = 0.f;
    for (int k = 0; k < kD; ++k) s += hrow[k] * W1[k * kD2 + m];
    s += b1[m];
    mid[m] = s > 0.f ? s : 0.f;
  }
  __syncthreads();
  for (int d = tid; d < kD; d += 256) {
    float s = 0.f;
    for (int k = 0; k < kD2; ++k) s += mid[k] * W2[k * kD + d];
    xr[d] = s + b2[d] + hrow[d];
  }
  __syncthreads();
  float ps = 0.f, pq = 0.f;
  for (int d = tid; d < kD; d += 256) { float x = xr[d]; ps += x; pq += x * x; }
  red[tid] = ps; __syncthreads();
  for (int s2 = 128; s2 > 0; s2 >>= 1) {
    if (tid < s2) red[tid] += red[tid + s2];
    __syncthreads();
  }
  const float mu = red[0] * (1.f / (float)kD);
  __syncthreads();
  red[tid] = pq; __syncthreads();
  for (int s2 = 128; s2 > 0; s2 >>= 1) {
    if (tid < s2) red[tid] += red[tid + s2];
    __syncthreads();
  }
  const float var = red[0] * (1.f / (float)kD) - mu * mu;
  const float rs  = rsqrtf(var + 1e-5f);
  for (int d = tid; d < kD; d += 256)
    hid_top[(b * kK + j) * kD + d] = (xr[d] - mu) * rs * gamma[d] + beta[d];
}

// ---------------------------------------------------------------------------
// Kernel 4: q = hid_last@Wr + br; scores over 16 slots (13 exact zeros);
// softmax; ctx = sum attn_j * hid_top_j
// ---------------------------------------------------------------------------
__global__ __launch_bounds__(256) void attend_kernel(
    const float* __restrict__ hidden_last, const float* __restrict__ Wr,
    const float* __restrict__ br, const float* __restrict__ hid_top,
    float* __restrict__ ctx)
{
  __shared__ float hl[kD];
  __shared__ float qv[kD];
  __shared__ float red[256];
  __shared__ float attn[kK];
  const int b = blockIdx.x, tid = threadIdx.x;
  for (int c = tid; c < kD; c += 256) hl[c] = hidden_last[b * kD + c];
  __syncthreads();
  for (int d = tid; d < kD; d += 256) {
    float s = 0.f;
    for (int k = 0; k < kD; ++k) s += hl[k] * Wr[k * kD + d];
    qv[d] = s + br[d];
  }
  __syncthreads();
  float sc[kK];
  for (int j = 0; j < kK; ++j) {
    float p = 0.f;
    for (int d = tid; d < kD; d += 256)
      p += hid_top[(b * kK + j) * kD + d] * qv[d];
    red[tid] = p; __syncthreads();
    for (int s2 = 128; s2 > 0; s2 >>= 1) {
      if (tid < s2) red[tid] += red[tid + s2];
      __syncthreads();
    }
    sc[j] = red[0];
    __syncthreads();
  }
  if (tid == 0) {
    float m = 0.f;                         // 13 slots have score exactly 0
    for (int j = 0; j < kK; ++j) m = fmaxf(m, sc[j]);
    float den = (float)(kSlots - kK) * __expf(-m);
    float e[kK];
    for (int j = 0; j < kK; ++j) { e[j] = __expf(sc[j] - m); den += e[j]; }
    for (int j = 0; j < kK; ++j) attn[j] = e[j] / den;
  }
  __syncthreads();
  for (int d = tid; d < kD; d += 256) {
    float s = 0.f;
    for (int j = 0; j < kK; ++j) s += attn[j] * hid_top[(b * kK + j) * kD + d];
    ctx[b * kD + d] = s;
  }
}

// ---------------------------------------------------------------------------
// Kernel 5: out(64 x 50257) = ctx(64x512) @ Wo(512x50257) + bo, fp32 WMMA.
// One wave per block; each wave computes TWO adjacent 16x16 N-tiles sharing
// one A fragment per k-step. grid = (ceil(V/32), 4 M-tiles).
// ---------------------------------------------------------------------------
__global__ __launch_bounds__(32) void out_gemm(
    const float* __restrict__ ctx, const float* __restrict__ Wo,
    const float* __restrict__ bo, float* __restrict__ out)
{
  const int lane  = threadIdx.x;
  const int n0    = blockIdx.x * 32;
  const int mt    = blockIdx.y;
  const int row16 = lane & 15;
  const int koff  = (lane >> 4) << 1;
  const int moff  = (lane >> 4) << 3;
  const int col0  = n0 + row16;
  const int col1  = col0 + 16;
  const bool cok0 = col0 < kV;
  const bool cok1 = col1 < kV;
  const int arow  = mt * 16 + row16;
  v8f acc0 = {0.f, 0.f, 0.f, 0.f, 0.f, 0.f, 0.f, 0.f};
  v8f acc1 = {0.f, 0.f, 0.f, 0.f, 0.f, 0.f, 0.f, 0.f};
  for (int k = 0; k < kD; k += 4) {
    v2f a;
    a.x = ctx[arow * kD + k + koff];
    a.y = ctx[arow * kD + k + koff + 1];
    const float* wr0 = &Wo[(long)(k + koff) * kV];
    const float* wr1 = &Wo[(long)(k + koff + 1) * kV];
    v2f b0, b1v;
    b0.x  = cok0 ? wr0[col0] : 0.f;
    b0.y  = cok0 ? wr1[col0] : 0.f;
    b1v.x = cok1 ? wr0[col1] : 0.f;
    b1v.y = cok1 ? wr1[col1] : 0.f;
    acc0 = wmma4(a, b0,  acc0);
    acc1 = wmma4(a, b1v, acc1);
  }
  if (cok0) {
    const float bias = bo[col0];
#pragma unroll
    for (int r = 0; r < 8; ++r)
      out[(long)(mt * 16 + r + moff) * kV + col0] = acc0[r] + bias;
  }
  if (cok1) {
    const float bias = bo[col1];
#pragma unroll
    for (int r = 0; r < 8; ++r)
      out[(long)(mt * 16 + r + moff) * kV + col1] = acc1[r] + bias;
  }
}

// ---------------------------------------------------------------------------
extern "C" void kernel_launch(void* const* d_in, const int* in_sizes, int n_in,
                              void* d_out, int out_size, void* d_ws, size_t ws_size,
                              hipStream_t stream) {
  const int*   seq         = (const int*)d_in[0];
  const float* temperature = (const float*)d_in[1];
  const float* gumbel      = (const float*)d_in[2];
  const float* embed       = (const float*)d_in[3];
  const float* W1          = (const float*)d_in[4];
  const float* b1          = (const float*)d_in[5];
  const float* W2          = (const float*)d_in[6];
  const float* b2          = (const float*)d_in[7];
  const float* gamma       = (const float*)d_in[8];
  const float* beta        = (const float*)d_in[9];
  const float* Wg          = (const float*)d_in[10];
  const float* bg          = (const float*)d_in[11];
  const float* Wr          = (const float*)d_in[12];
  const float* br          = (const float*)d_in[13];
  const float* Wo          = (const float*)d_in[14];
  const float* bo          = (const float*)d_in[15];
  float* out = (float*)d_out;

  // Workspace layout (~1.2 MB total)
  float* ws          = (float*)d_ws;
  float* gate        = ws;                       // B * 2048
  float* hidden_last = gate + kB * kTC;          // B * 512
  int*   topidx      = (int*)(hidden_last + kB * kD);  // B*K (padded to 256)
  float* hid_top     = (float*)(topidx + 256);   // B * K * 512
  float* ctx         = hid_top + kB * kK * kD;   // B * 512

  const size_t smem1 = (size_t)(16 * HS + 16 * AS) * sizeof(float); // ~97 KB

  ffn_ln_gate<<<dim3((kB * kTF) / 16), dim3(256), smem1, stream>>>(
      seq, embed, W1, b1, W2, b2, gamma, beta, Wg, bg, gate, hidden_last);

  topk_kernel<<<dim3(kB), dim3(256), 0, stream>>>(
      gate, gumbel, temperature, topidx);

  recompute_hidden<<<dim3(kB * kK), dim3(256), 0, stream>>>(
      seq, embed, W1, b1, W2, b2, gamma, beta, topidx, hid_top);

  attend_kernel<<<dim3(kB), dim3(256), 0, stream>>>(
      hidden_last, Wr, br, hid_top, ctx);

  out_gemm<<<dim3((kV + 31) / 32, 4), dim3(32), 0, stream>>>(
      ctx, Wo, bo, out);
}